// XYZ_86071144612333
// MI455X (gfx1250) — compile-verified
//
#include <hip/hip_runtime.h>
#include <stdint.h>

// ---------------------------------------------------------------------------
// Problem geometry (from reference): data (128, 2, 72, 2084) f32
// out (128, 4, 72, 2084) f32.  Streaming elementwise op -> HBM-bound
// (~460 MB traffic, ~20us floor @ 23.3 TB/s).  Strategy:
//   * thread owns one (y, 4x) group, loops over a chunk of batches so the
//     10 trig evaluations amortize across the batch loop
//   * gfx1250 async path: global_load_async_to_lds_b128 double-buffer for the
//     dist/mask tiles, synchronized with s_wait_asynccnt (ASYNCcnt)
//   * B128 loads from LDS, non-temporal B128 stores to global (write-once
//     stream larger than the 192MB L2)
// ---------------------------------------------------------------------------

typedef float v4f __attribute__((ext_vector_type(4)));

#define NB        128
#define YS        72
#define XS        2084
#define XQ        (XS / 4)        // 521 float4 groups per row
#define NP        (YS * XQ)       // 37512 point-groups
#define SLICE     (YS * XS)       // 150048 elements per (b, channel)
#define BCHUNK    8
#define NBCHUNKS  (NB / BCHUNK)   // 16
#define TPB       256

#define DEG2RAD   0.017453292519943295  // pi/180, double

__global__ __launch_bounds__(TPB) void xyz_project_kernel(
    const float* __restrict__ data, float* __restrict__ out)
{
  // double buffer: [buf][channel(dist,mask)][thread][4 floats] = 16 KB
  __shared__ __attribute__((aligned(16))) float smem[2][2][TPB][4];

  const int t = threadIdx.x;
  const int p = blockIdx.x * TPB + t;
  if (p >= NP) return;

  const int y  = p / XQ;
  const int xq = p - y * XQ;
  const int x0 = xq * 4;

  // ---- vertical angle: two linspace segments, numpy-faithful in double ----
  double adeg;
  if (y < 40) {
    const double start = 13.0 / 3.0;            //  4 + 1/3
    const double stop  = -25.0 / 3.0;           // -8 - 1/3
    adeg = (y == 39) ? stop : start + (double)y * ((stop - start) / 39.0);
  } else {
    const int i = y - 40;
    const double start = -53.0 / 6.0;           // -8 - 1/3 - 1/2
    const double stop  = -73.0 / 3.0;           // -24 - 1/3
    adeg = (i == 31) ? stop : start + (double)i * ((stop - start) / 31.0);
  }
  const float av = (float)(adeg * DEG2RAD);
  const float cv = cosf(av);
  const float pz = sinf(av);

  // ---- horizontal angles for the 4 x's of this group ----
  float px[4], py[4];
#pragma unroll
  for (int k = 0; k < 4; ++k) {
    const double hdeg = (double)(2083 - (x0 + k)) * 0.1728 + 180.0;
    const float ah = (float)(hdeg * DEG2RAD);
    px[k] = cosf(ah) * cv;
    py[k] = sinf(ah) * cv;
  }

  const int    b0  = blockIdx.y * BCHUNK;
  const size_t row = (size_t)y * XS + x0;       // element offset within a slice

  // LDS byte offsets for this thread's staging slots (generic ptr low 32 bits
  // are the LDS offset per the gfx1250 aperture scheme)
  const uint32_t ldsD[2] = { (uint32_t)(uintptr_t)&smem[0][0][t][0],
                             (uint32_t)(uintptr_t)&smem[1][0][t][0] };
  const uint32_t ldsM[2] = { (uint32_t)(uintptr_t)&smem[0][1][t][0],
                             (uint32_t)(uintptr_t)&smem[1][1][t][0] };

  // async issue: stage dist+mask float4 for batch b into LDS buffer `buf`
  auto issue = [&](int b, int buf) {
    const uint64_t ad = (uint64_t)(uintptr_t)(data + (size_t)b * 2 * SLICE + row);
    const uint64_t am = ad + (uint64_t)SLICE * sizeof(float);
    asm volatile("global_load_async_to_lds_b128 %0, %1, off"
                 :: "v"(ldsD[buf]), "v"(ad) : "memory");
    asm volatile("global_load_async_to_lds_b128 %0, %1, off"
                 :: "v"(ldsM[buf]), "v"(am) : "memory");
  };

  issue(b0, 0);

  for (int i = 0; i < BCHUNK; ++i) {
    const int buf = i & 1;
    if (i + 1 < BCHUNK) {
      issue(b0 + i + 1, buf ^ 1);
      // 4 async loads outstanding; wait until only the prefetched pair remains
      asm volatile("s_wait_asynccnt 0x2" ::: "memory");
    } else {
      asm volatile("s_wait_asynccnt 0x0" ::: "memory");
    }

    const v4f dd = *(const v4f*)&smem[buf][0][t][0];
    const v4f mm = *(const v4f*)&smem[buf][1][t][0];

    v4f ox, oy, oz;
#pragma unroll
    for (int k = 0; k < 4; ++k) {
      const bool on = (mm[k] >= 0.5f);
      const float d = dd[k];
      ox[k] = on ? d * px[k] : 0.0f;
      oy[k] = on ? d * py[k] : 0.0f;
      oz[k] = on ? d * pz    : 0.0f;
    }

    float* ob = out + (size_t)(b0 + i) * 4 * SLICE + row;
    __builtin_nontemporal_store(ox, (v4f*)(ob));
    __builtin_nontemporal_store(oy, (v4f*)(ob + (size_t)SLICE));
    __builtin_nontemporal_store(oz, (v4f*)(ob + (size_t)2 * SLICE));
    __builtin_nontemporal_store(mm, (v4f*)(ob + (size_t)3 * SLICE));
  }
}

extern "C" void kernel_launch(void* const* d_in, const int* in_sizes, int n_in,
                              void* d_out, int out_size, void* d_ws, size_t ws_size,
                              hipStream_t stream) {
  (void)in_sizes; (void)n_in; (void)out_size; (void)d_ws; (void)ws_size;
  const float* data = (const float*)d_in[0];
  float* out = (float*)d_out;

  dim3 grid((NP + TPB - 1) / TPB, NBCHUNKS, 1);
  dim3 block(TPB, 1, 1);
  hipLaunchKernelGGL(xyz_project_kernel, grid, block, 0, stream, data, out);
}